// RandomMFGL_82480551952956
// MI455X (gfx1250) — compile-verified
//
#include <hip/hip_runtime.h>

typedef float v2f __attribute__((ext_vector_type(2)));
typedef float v8f __attribute__((ext_vector_type(8)));

#define N_ENS 4
#define N_IN  16384
#define N_OUT 4096
#define IN_C  16
#define OUT_C 16

// Block: 512 threads = 16 wave32s. wave w: ensemble n = w>>2, K-slice = w&3.
// Grid: N_OUT/16 = 256 blocks, one 16-row output tile each.
__global__ __launch_bounds__(512, 1)
void fgl_wmma_kernel(const float* __restrict__ x,
                     const float* __restrict__ As,
                     const float* __restrict__ Ws,
                     const float* __restrict__ bs,
                     float* __restrict__ out)
{
    __shared__ float lds_part[16 * 256];   // per-wave partial C tiles
    __shared__ float lds_h[N_ENS * 256];   // k-reduced h_n tiles

    const int tid  = threadIdx.x;
    const int lane = tid & 31;
    const int w    = tid >> 5;
    const int n    = w >> 2;        // ensemble member
    const int ks   = w & 3;         // K-slice index
    const int ot   = blockIdx.x;    // output row tile

    const int m  = lane & 15;       // A: row M; B/C: column N
    const int hh = lane >> 4;       // half-wave selects K pair
    const int kb = ks * (N_IN / 4);

    // A operand: lane m streams row (ot*16+m) of A_n; halves offset by 2 in K.
    const float* rowA = As + (size_t)n * N_OUT * N_IN
                           + (size_t)(ot * 16 + m) * N_IN
                           + 2 * hh;
    // B operand: lane holds column m of x; halves offset by 2 rows in K.
    const float* colX = x + 2 * hh * IN_C + m;

    // -------- software-pipelined streaming WMMA loop --------
    constexpr int ST    = 8;              // pipeline depth (distinct reg stages)
    constexpr int STEPS = (N_IN / 4) / 4; // 1024 K-steps of 4 per wave

    v2f abuf[ST];
    v2f bbuf[ST];

#pragma unroll
    for (int s = 0; s < ST; ++s) {
        const int k = kb + 4 * s;
        abuf[s] = __builtin_nontemporal_load((const v2f*)(rowA + k));
        bbuf[s].x = colX[k * IN_C];
        bbuf[s].y = colX[k * IN_C + IN_C];
    }

    v8f acc = {};
    for (int it = 0; it < STEPS - ST; it += ST) {
#pragma unroll
        for (int s = 0; s < ST; ++s) {
            acc = __builtin_amdgcn_wmma_f32_16x16x4_f32(
                false, abuf[s], false, bbuf[s], (short)0, acc, false, false);
            const int k = kb + 4 * (it + ST + s);
            abuf[s] = __builtin_nontemporal_load((const v2f*)(rowA + k));
            bbuf[s].x = colX[k * IN_C];
            bbuf[s].y = colX[k * IN_C + IN_C];
        }
    }
#pragma unroll
    for (int s = 0; s < ST; ++s) {
        acc = __builtin_amdgcn_wmma_f32_16x16x4_f32(
            false, abuf[s], false, bbuf[s], (short)0, acc, false, false);
    }

    // Spill partial C: lane holds C[r + 8*hh][m] in acc[r]
#pragma unroll
    for (int r = 0; r < 8; ++r)
        lds_part[w * 256 + (r + 8 * hh) * 16 + m] = acc[r];
    __syncthreads();

    // Reduce the 4 K-slice partials per ensemble (1024 sums, 2 per thread)
    for (int e = tid; e < N_ENS * 256; e += 512) {
        const int en  = e >> 8;
        const int idx = e & 255;
        float s = lds_part[(en * 4 + 0) * 256 + idx]
                + lds_part[(en * 4 + 1) * 256 + idx]
                + lds_part[(en * 4 + 2) * 256 + idx]
                + lds_part[(en * 4 + 3) * 256 + idx];
        lds_h[en * 256 + idx] = s;
    }
    __syncthreads();

    // y[m][g] = 0.25 * sum_n ( sum_f h_n[m][f] * W_n[f][g] + b_n[g] )
    if (tid < 256) {
        const int om = tid >> 4;
        const int g  = tid & 15;
        float accy = 0.f;
#pragma unroll
        for (int en = 0; en < N_ENS; ++en) {
            float s = 0.f;
#pragma unroll
            for (int f = 0; f < IN_C; ++f)
                s += lds_h[en * 256 + om * 16 + f]
                   * Ws[en * IN_C * OUT_C + f * OUT_C + g];
            accy += s + bs[en * OUT_C + g];
        }
        out[(ot * 16 + om) * OUT_C + g] = accy * 0.25f;
    }
}

extern "C" void kernel_launch(void* const* d_in, const int* in_sizes, int n_in,
                              void* d_out, int out_size, void* d_ws, size_t ws_size,
                              hipStream_t stream) {
    const float* x  = (const float*)d_in[0];   // [1, 16384, 16]
    const float* As = (const float*)d_in[1];   // [4, 4096, 16384]
    const float* Ws = (const float*)d_in[2];   // [4, 16, 16]
    const float* bs = (const float*)d_in[3];   // [4, 16]
    float* out = (float*)d_out;                // [1, 4096, 16]

    dim3 grid(N_OUT / 16);
    dim3 block(512);
    hipLaunchKernelGGL(fgl_wmma_kernel, grid, block, 0, stream,
                       x, As, Ws, bs, out);
}